// RefWignerRotationZ_7232724927046
// MI455X (gfx1250) — compile-verified
//
#include <hip/hip_runtime.h>

// ---------------------------------------------------------------------------
// Wigner-Z rotation: out[b,d,f] = cos(a_b*freq_d)*x[b,d,f] + sin(a_b*freq_d)*x[b,rev_d,f]
// x: [2048, 592, 64] f32.  Memory-bound (~621 MB @ 23.3 TB/s => ~27 us floor).
// Strategy: async global->LDS staging (ASYNCcnt), LDS gather for the mirror,
// b128 everywhere, non-temporal b128 stores.
// ---------------------------------------------------------------------------

#define DIMR 592
#define BATCH 2048
#define F4 16                       // float4 per row (64 floats)
#define ROWS_PER_TILE 74
#define NTILES 8                    // 8 * 74 = 592
#define HALO 6                      // max mirror distance (l=3 block: 7 rows)
#define TILE_ROWS (ROWS_PER_TILE + 2 * HALO)   // 86
#define TILE_V4 (TILE_ROWS * F4)               // 1376  (22016 B LDS)
#define COMP_V4 (ROWS_PER_TILE * F4)           // 1184
#define TPB 256

typedef float f32x4 __attribute__((ext_vector_type(4)));
typedef int   i32x4 __attribute__((ext_vector_type(4)));

// address-space-qualified pointer types matching the async-LDS builtin signature
typedef __attribute__((address_space(1))) i32x4 g_i32x4;
typedef __attribute__((address_space(3))) i32x4 l_i32x4;

// ---- compile-time freq / rev tables (mirrors the Python _build_tables) ----
struct Tables {
  float freq[DIMR];
  short rev[DIMR];
};

static constexpr Tables make_tables() {
  Tables t{};
  const int L[4]   = {0, 1, 2, 3};
  const int MUL[4] = {128, 64, 32, 16};
  int off = 0;
  for (int k = 0; k < 4; ++k) {
    const int l = L[k], d = 2 * l + 1;
    for (int m = 0; m < MUL[k]; ++m) {
      for (int i = 0; i < d; ++i) {
        t.freq[off + i] = (float)(l - i);            // m = l .. -l
        t.rev[off + i]  = (short)(off + d - 1 - i);  // mirrored index
      }
      off += d;
    }
  }
  return t;
}

__constant__ Tables c_tab = make_tables();

// ---- CDNA5 async global->LDS helpers (builtin if available, else asm) ----
__device__ __forceinline__ void async_copy_b128(const f32x4* gbase,
                                                unsigned byte_off,
                                                f32x4* lds_dst) {
#if __has_builtin(__builtin_amdgcn_global_load_async_to_lds_b128)
  __builtin_amdgcn_global_load_async_to_lds_b128(
      (g_i32x4*)((unsigned long long)gbase + (unsigned long long)byte_off),
      (l_i32x4*)(void*)lds_dst, 0, 0);
#else
  const unsigned lds_addr = (unsigned)(unsigned long long)(void*)lds_dst;
  asm volatile("global_load_async_to_lds_b128 %0, %1, %2"
               :
               : "v"(lds_addr), "v"(byte_off),
                 "s"((unsigned long long)gbase)
               : "memory");
#endif
}

__device__ __forceinline__ void wait_async_zero() {
#if __has_builtin(__builtin_amdgcn_s_wait_asynccnt)
  __builtin_amdgcn_s_wait_asynccnt(0);
#else
  asm volatile("s_wait_asynccnt 0" ::: "memory");
#endif
}

// ---------------------------------------------------------------------------
__global__ void __launch_bounds__(TPB)
wigner_z_kernel(const float* __restrict__ x, const float* __restrict__ angle,
                float* __restrict__ out) {
  __shared__ f32x4 tile[TILE_V4];

  const int tid = threadIdx.x;
  const int b   = blockIdx.y;
  const int r0  = blockIdx.x * ROWS_PER_TILE;

  const float a = angle[b];  // uniform -> scalar load

  const f32x4* xv = (const f32x4*)x;
  f32x4* ov       = (f32x4*)out;
  const unsigned long long base_row = (unsigned long long)b * DIMR;

  // ---- stage tile (+halo) into LDS via async b128 copies ----
  for (int i = tid; i < TILE_V4; i += TPB) {
    int gr = r0 - HALO + (i >> 4);
    gr = gr < 0 ? 0 : (gr > DIMR - 1 ? DIMR - 1 : gr);  // clamp: never read anyway
    const unsigned byte_off =
        (unsigned)(((base_row + (unsigned long long)gr) * F4 + (i & 15)) *
                   sizeof(f32x4));
    async_copy_b128(xv, byte_off, &tile[i]);
  }
  wait_async_zero();   // this wave's async copies landed in LDS
  __syncthreads();     // all waves' copies visible

  // ---- rotate: one float4 per thread-iteration ----
  for (int j = tid; j < COMP_V4; j += TPB) {
    const int dl = j >> 4;      // row within tile
    const int f  = j & 15;      // float4 within row
    const int d  = r0 + dl;     // row within DIMR

    float s, c;
    __sincosf(a * c_tab.freq[d], &s, &c);

    const f32x4 xo = tile[(dl + HALO) * F4 + f];
    const int   rl = (int)c_tab.rev[d] - r0 + HALO;  // in [0, TILE_ROWS)
    const f32x4 xr = tile[rl * F4 + f];

    const f32x4 r = xo * c + xr * s;   // contracts to v_fma per lane

    __builtin_nontemporal_store(r, &ov[(base_row + (unsigned long long)d) * F4 + f]);
  }
}

// ---------------------------------------------------------------------------
extern "C" void kernel_launch(void* const* d_in, const int* in_sizes, int n_in,
                              void* d_out, int out_size, void* d_ws, size_t ws_size,
                              hipStream_t stream) {
  (void)in_sizes; (void)n_in; (void)out_size; (void)d_ws; (void)ws_size;
  const float* x     = (const float*)d_in[0];
  const float* angle = (const float*)d_in[1];
  float* out         = (float*)d_out;

  dim3 grid(NTILES, BATCH, 1);
  wigner_z_kernel<<<grid, TPB, 0, stream>>>(x, angle, out);
}